// InterferenceAttention_59734405153517
// MI455X (gfx1250) — compile-verified
//
#include <hip/hip_runtime.h>
#include <math.h>

#define D_MODEL   512
#define NUM_HEADS 8
#define NUM_WAVES 16
#define HEAD_DIM  64
#define T_SEQ     512
#define BATCH     2
#define EPSV      1e-8f
#define MBLK      128          // 8 waves x 16 rows per block
#define KC        128          // K chunk staged in LDS per step
#define APAD      4            // LDS row padding (dwords) -> conflict-free ds_load_b64

typedef __attribute__((ext_vector_type(2))) float v2f;
typedef __attribute__((ext_vector_type(8))) float v8f;
typedef __attribute__((ext_vector_type(4))) unsigned int v4u;
typedef __attribute__((ext_vector_type(4))) int v4i;
typedef __attribute__((ext_vector_type(8))) int v8i;

// ---------------------------------------------------------------------------
// TDM 2-D f32 tile load: tile_w contiguous elements per row, tile_h rows,
// row stride `stride_elems`, optional LDS padding (pad_amount+1 dwords every
// 2^(pad_interval+1) dwords). Descriptor per CDNA5 ISA §8.3-8.6.
// ---------------------------------------------------------------------------
__device__ __forceinline__ void tdm_load_2d_f32(const float* gsrc, int stride_elems,
                                                int tile_w, int tile_h,
                                                unsigned lds_byte_addr,
                                                int pad_en, int pad_iv, int pad_amt) {
  unsigned long long ga = (unsigned long long)(size_t)gsrc;
  v4u g0;
  g0[0] = 1u;                                             // count=1, user descriptor
  g0[1] = lds_byte_addr;                                  // lds_addr
  g0[2] = (unsigned)(ga & 0xFFFFFFFFu);                   // global_addr[31:0]
  g0[3] = (unsigned)((ga >> 32) & 0x01FFFFFFu)            // global_addr[56:32]
          | (2u << 30);                                   // type=2 ("image")
  const unsigned tw = (unsigned)tile_w, th = (unsigned)tile_h;
  v8i g1;
  g1[0] = (int)((2u << 16) |                              // data_size = 4B
                ((unsigned)pad_en  << 20) |
                ((unsigned)pad_iv  << 22) |
                ((unsigned)pad_amt << 25));
  g1[1] = (int)((tw & 0xFFFFu) << 16);                    // tensor_dim0[15:0]
  g1[2] = (int)(((tw >> 16) & 0xFFFFu) |                  // tensor_dim0[31:16]
                ((th & 0xFFFFu) << 16));                  // tensor_dim1[15:0]
  g1[3] = (int)(((th >> 16) & 0xFFFFu) |                  // tensor_dim1[31:16]
                ((tw & 0xFFFFu) << 16));                  // tile_dim0 = tile_w
  g1[4] = (int)(th & 0xFFFFu);                            // tile_dim1 = tile_h
  g1[5] = stride_elems;                                   // tensor_dim0_stride[31:0]
  g1[6] = 0;
  g1[7] = 0;
  v4i z4 = {0, 0, 0, 0};
  v8i z8 = {0, 0, 0, 0, 0, 0, 0, 0};
  __builtin_amdgcn_tensor_load_to_lds(g0, g1, z4, z4, z8, 0);
}

// ---------------------------------------------------------------------------
// Batched GEMM: one 128x16 output block per 256-thread block (8 waves, one
// 16x16 tile each, stacked in M). Per K-chunk, the TDM stages the 128xKC A
// panel (padded rows: stride KC+APAD) and the KCx16 B panel into LDS; waves
// then run a pure LDS -> v_wmma_f32_16x16x4_f32 inner loop with full EXEC.
// WMMA VGPR layouts per CDNA5 ISA §7.12.2.
// ---------------------------------------------------------------------------
__global__ void gemm_wmma_f32(const float* __restrict__ A, const float* __restrict__ B,
                              const float* __restrict__ bias, float* __restrict__ D,
                              int M, int N, int K, int lda, int ldb, int ldd,
                              long sAy, long sAz, long sBy, long sBz, long sDy, long sDz,
                              int epi, int causal) {
  __shared__ float As[MBLK * (KC + APAD)];   // 66 KB
  __shared__ float Bs[KC * 16];              //  8 KB

  const int ntn = N >> 4;
  const int nt  = blockIdx.x % ntn;                       // column tile
  const int mb  = blockIdx.x / ntn;                       // 128-row super-block
  const int wid = __builtin_amdgcn_readfirstlane(threadIdx.x >> 5);  // SGPR wave id
  const int mtile = mb * (MBLK / 16) + wid;               // scalar

  const float* Ab = A + (size_t)blockIdx.y * sAy + (size_t)blockIdx.z * sAz;
  const float* Bb = B + (size_t)blockIdx.y * sBy + (size_t)blockIdx.z * sBz;
  float*       Db = D + (size_t)blockIdx.y * sDy + (size_t)blockIdx.z * sDz;

  // Causal (attn@v): rows q in this wave's tile need only k <= q.
  const int KeffW  = causal ? ((mtile + 1) * 16 < K ? (mtile + 1) * 16 : K) : K;
  // Block-uniform chunk count (max over the 8 waves of this block).
  const int Kblock = causal ? ((mb + 1) * MBLK < K ? (mb + 1) * MBLK : K) : K;
  const int nch    = Kblock / KC;

  const int lane = threadIdx.x & 31;
  const int half = lane >> 4;
  const int idx  = lane & 15;

  v8f acc = {};
  const float* arowL = As + (size_t)(16 * wid + idx) * (KC + APAD) + 2 * half;
  const float* browL = Bs + 2 * half * 16 + idx;

  for (int c = 0; c < nch; ++c) {
    const int lo = c * KC;
    if (wid == 0) {
      // A chunk: 128 rows x KC, padded to KC+APAD dword rows in LDS
      // (pad_interval=6 -> every 128 dwords, pad_amount=3 -> 4 dwords).
      tdm_load_2d_f32(Ab + (size_t)(mb * MBLK) * lda + lo, lda, KC, MBLK,
                      (unsigned)(size_t)(void*)As, 1, 6, 3);
      // B chunk: KC rows x 16 cols of this column tile.
      tdm_load_2d_f32(Bb + (size_t)lo * ldb + nt * 16, ldb, 16, KC,
                      (unsigned)(size_t)(void*)Bs, 0, 0, 0);
      __builtin_amdgcn_s_wait_tensorcnt(0);
    }
    __syncthreads();

    int kcEff = KeffW - lo;                 // scalar; may be <=0 for causal waves
    kcEff = kcEff < 0 ? 0 : (kcEff > KC ? KC : kcEff);
#pragma unroll 4
    for (int kb = 0; kb < kcEff; kb += 4) {
      v2f a = *(const v2f*)(arowL + kb);    // ds_load_b64: A[m][lo+kb+2h .. +1]
      v2f b;
      b.x = browL[kb * 16];                 // Bs[(kb+2h)  ][n]
      b.y = browL[kb * 16 + 16];            // Bs[(kb+2h+1)][n]
      acc = __builtin_amdgcn_wmma_f32_16x16x4_f32(false, a, false, b,
                                                  (short)0, acc, false, false);
    }
    __syncthreads();
  }

  const float bv = bias ? bias[nt * 16 + idx] : 0.0f;
#pragma unroll
  for (int j = 0; j < 8; ++j) {
    const int m = mtile * 16 + j + 8 * half;
    float val = acc[j] + bv;
    if (epi == 1) {                        // softplus + EPS (amplitude path)
      val = (val > 20.0f) ? val : log1pf(expf(val));
      val += EPSV;
    }
    Db[(size_t)m * ldd + nt * 16 + idx] = val;
  }
}

// ---------------------------------------------------------------------------
// Interference intensity -> normalized attention. One block per (b,h,q) row;
// q-row wave params in LDS; threads sweep k with float4 parameter loads.
// ---------------------------------------------------------------------------
__global__ void interference_attn_kernel(const float* __restrict__ F,
                                         const float* __restrict__ P,
                                         const float* __restrict__ Aamp,
                                         float* __restrict__ attn) {
  const int q  = blockIdx.x % T_SEQ;
  const int bh = blockIdx.x / T_SEQ;
  const int h  = bh % NUM_HEADS;
  const int b  = bh / NUM_HEADS;

  __shared__ float fq[NUM_WAVES], pq[NUM_WAVES], aq[NUM_WAVES];
  const int tid = threadIdx.x;
  const size_t rowq = (size_t)(b * T_SEQ + q) * (NUM_HEADS * NUM_WAVES) + h * NUM_WAVES;
  if (tid < NUM_WAVES)            fq[tid]      = F[rowq + tid];
  else if (tid < 2 * NUM_WAVES)   pq[tid - 16] = P[rowq + tid - 16];
  else if (tid < 3 * NUM_WAVES)   aq[tid - 32] = Aamp[rowq + tid - 32];
  __syncthreads();

  float* arow = attn + ((size_t)bh * T_SEQ + q) * T_SEQ;
  for (int k = tid; k < T_SEQ; k += blockDim.x) {
    float val = 0.0f;
    if (k <= q) {
      const size_t rowk = (size_t)(b * T_SEQ + k) * (NUM_HEADS * NUM_WAVES) + h * NUM_WAVES;
      const float dt = (float)(q - k);
      float inten = 0.0f, energy = 0.0f;
      const float4* f4 = (const float4*)(F + rowk);      // rows are 64B-aligned
      const float4* p4 = (const float4*)(P + rowk);
      const float4* a4 = (const float4*)(Aamp + rowk);
#pragma unroll
      for (int v = 0; v < NUM_WAVES / 4; ++v) {
        const float4 fk = f4[v], pk = p4[v], ak = a4[v];
        const float* fkp = (const float*)&fk;
        const float* pkp = (const float*)&pk;
        const float* akp = (const float*)&ak;
#pragma unroll
        for (int j = 0; j < 4; ++j) {
          const int w = v * 4 + j;
          const float Aq = aq[w], Ak = akp[j];
          const float arg = (fq[w] - fkp[j]) * dt + (pq[w] - pkp[j]);
          inten  += Aq * Aq + Ak * Ak + 2.0f * Aq * Ak * cosf(arg);
          const float s = Aq + Ak;
          energy += s * s;
        }
      }
      val = inten / (energy + EPSV);
    }
    arow[k] = val;
  }
}

// ---------------------------------------------------------------------------
extern "C" void kernel_launch(void* const* d_in, const int* in_sizes, int n_in,
                              void* d_out, int out_size, void* d_ws, size_t ws_size,
                              hipStream_t stream) {
  const float* x  = (const float*)d_in[0];
  const float* Wf = (const float*)d_in[1];
  const float* bf = (const float*)d_in[2];
  const float* Wp = (const float*)d_in[3];
  const float* bp = (const float*)d_in[4];
  const float* Wa = (const float*)d_in[5];
  const float* ba = (const float*)d_in[6];
  const float* Wv = (const float*)d_in[7];
  const float* bv = (const float*)d_in[8];
  const float* Wo = (const float*)d_in[9];
  const float* bo = (const float*)d_in[10];
  float* out = (float*)d_out;

  const int BT = BATCH * T_SEQ;          // 1024 rows of x
  const int HW = NUM_HEADS * NUM_WAVES;  // 128

  // Workspace (f32): F,P,A (1024x128), V (1024x512), attn (16x512x512), H (1024x512)
  float* Fb = (float*)d_ws;
  float* Pb = Fb + (size_t)BT * HW;
  float* Ab = Pb + (size_t)BT * HW;
  float* Vb = Ab + (size_t)BT * HW;
  float* At = Vb + (size_t)BT * D_MODEL;
  float* Hb = At + (size_t)BATCH * NUM_HEADS * T_SEQ * T_SEQ;

  const dim3 blk(256);
  auto nblk = [](int M, int N) { return (M / MBLK) * (N / 16); };

  // Stage 1: projections (f32 WMMA, TDM-staged A/B panels)
  gemm_wmma_f32<<<dim3(nblk(BT, HW)), blk, 0, stream>>>(
      x, Wf, bf, Fb, BT, HW, D_MODEL, D_MODEL, HW, HW, 0, 0, 0, 0, 0, 0, 0, 0);
  gemm_wmma_f32<<<dim3(nblk(BT, HW)), blk, 0, stream>>>(
      x, Wp, bp, Pb, BT, HW, D_MODEL, D_MODEL, HW, HW, 0, 0, 0, 0, 0, 0, 0, 0);
  gemm_wmma_f32<<<dim3(nblk(BT, HW)), blk, 0, stream>>>(
      x, Wa, ba, Ab, BT, HW, D_MODEL, D_MODEL, HW, HW, 0, 0, 0, 0, 0, 0, 1, 0);
  gemm_wmma_f32<<<dim3(nblk(BT, D_MODEL)), blk, 0, stream>>>(
      x, Wv, bv, Vb, BT, D_MODEL, D_MODEL, D_MODEL, D_MODEL, D_MODEL,
      0, 0, 0, 0, 0, 0, 0, 0);

  // Stage 2: interference intensity -> attention (VALU/cos bound, L2-resident)
  interference_attn_kernel<<<dim3(BATCH * NUM_HEADS * T_SEQ), blk, 0, stream>>>(
      Fb, Pb, Ab, At);

  // Stage 3: out_h = attn @ v  (causal: only mb+1 K-chunks staged/computed)
  gemm_wmma_f32<<<dim3(nblk(T_SEQ, HEAD_DIM), NUM_HEADS, BATCH), blk, 0, stream>>>(
      At, Vb, nullptr, Hb, T_SEQ, HEAD_DIM, T_SEQ, T_SEQ, D_MODEL, D_MODEL,
      (long)T_SEQ * T_SEQ, (long)NUM_HEADS * T_SEQ * T_SEQ,
      (long)HEAD_DIM,      (long)T_SEQ * D_MODEL,
      (long)HEAD_DIM,      (long)T_SEQ * D_MODEL, 0, 1);

  // Stage 4: final output projection
  gemm_wmma_f32<<<dim3(nblk(BT, D_MODEL)), blk, 0, stream>>>(
      Hb, Wo, bo, out, BT, D_MODEL, D_MODEL, D_MODEL, D_MODEL, D_MODEL,
      0, 0, 0, 0, 0, 0, 0, 0);
}